// GradientProcessor_58634893525577
// MI455X (gfx1250) — compile-verified
//
#include <hip/hip_runtime.h>

#define CROP  128
#define PH    300
#define PW    300
#define IMG_H 640
#define IMG_W 640
#define CCH   3
#define NB    8
#define NP    16

typedef unsigned int v4u __attribute__((ext_vector_type(4)));
typedef int          v4i __attribute__((ext_vector_type(4)));
typedef int          v8i __attribute__((ext_vector_type(8)));

// ---------------------------------------------------------------------------
// Kernel 1: analytic grad of TV loss; initializes d_out.
//   d1(i,j) = p(i,j) - p(i,j+1)  (0 on last col)
//   d2(i,j) = p(i,j) - p(i+1,j)  (0 on last row)
//   r = sqrt(d1^2 + d2^2)
//   g(i,j) = (d1+d2)/r |(i,j)  -  d1/r |(i,j-1)  -  d2/r |(i-1,j)
// ---------------------------------------------------------------------------
__global__ void tv_grad_kernel(const float* __restrict__ patch,
                               float* __restrict__ out) {
  int idx = blockIdx.x * blockDim.x + threadIdx.x;
  if (idx >= PH * PW * CCH) return;
  int c = idx % CCH;
  int j = (idx / CCH) % PW;
  int i = idx / (CCH * PW);

  auto PV = [&](int y, int x) { return patch[(y * PW + x) * CCH + c]; };
  auto TERM = [&](int y, int x, float& d1, float& d2, float& r) {
    float ctr = PV(y, x);
    d1 = (x < PW - 1) ? (ctr - PV(y, x + 1)) : 0.f;
    d2 = (y < PH - 1) ? (ctr - PV(y + 1, x)) : 0.f;
    r = sqrtf(d1 * d1 + d2 * d2);
  };

  float d1, d2, r, g = 0.f;
  TERM(i, j, d1, d2, r);
  if (r > 0.f) g += (d1 + d2) / r;
  if (j > 0) { TERM(i, j - 1, d1, d2, r); if (r > 0.f) g -= d1 / r; }
  if (i > 0) { TERM(i - 1, j, d1, d2, r); if (r > 0.f) g -= d2 / r; }
  out[idx] = g;
}

// ---------------------------------------------------------------------------
// Kernel 2: one workgroup per (b,p). TDM DMA of the 128x128x3 crop into LDS
// (196608 B -- needs CDNA5's 320KB-per-WGP LDS), then bilinear resample with
// the flip/rot transforms folded into the sampling coordinate, atomic-add
// into the 300x300x3 accumulator.
// ---------------------------------------------------------------------------
__global__ void patch_accum_kernel(const float* __restrict__ grads,
                                   const int* __restrict__ boxes,
                                   const unsigned char* __restrict__ decs,
                                   float* __restrict__ out) {
  extern __shared__ float tile[];               // [CROP][CROP][CCH]

  const int bp   = blockIdx.x;                  // bp = b*16 + p
  const int ymin = boxes[bp * 4 + 0];
  const int xmin = boxes[bp * 4 + 1];
  const int dec0 = decs[bp * 3 + 0];            // rot90 k=3 (applied last)
  const int dec1 = decs[bp * 3 + 1];            // flip left-right
  const int dec2 = decs[bp * 3 + 2];            // flip up-down (applied first)

  // ---- wave 0 issues the Tensor Data Mover load, waits TENSORcnt==0 ----
  if (threadIdx.x == 0) {
    unsigned long long gaddr =
        (unsigned long long)grads +
        4ull * (unsigned long long)((((bp >> 4) * IMG_H + ymin) * IMG_W + xmin) * CCH);
    // generic shared pointer: low 32 bits are the LDS byte offset
    unsigned ldsoff = (unsigned)(unsigned long long)(void*)tile;

    v4u g0;
    g0.x = 1u;                                   // count=1 (valid), no gather
    g0.y = ldsoff;                               // lds_addr
    g0.z = (unsigned)(gaddr & 0xFFFFFFFFu);      // global_addr[31:0]
    g0.w = (unsigned)(gaddr >> 32) | 0x80000000u;// global_addr[56:32] | type=2

    v8i g1;
    g1[0] = (int)(2u << 16);                     // data_size=2 (4 bytes)
    g1[1] = (int)((384u & 0xFFFFu) << 16);       // tensor_dim0[15:0] @ bits48..63
    g1[2] = (int)(128u << 16);                   // tensor_dim0 hi=0 | tensor_dim1 lo=128
    g1[3] = (int)(384u << 16);                   // tensor_dim1 hi=0 | tile_dim0=384
    g1[4] = 128;                                 // tile_dim1=128, tile_dim2=0
    g1[5] = IMG_W * CCH;                         // tensor_dim0_stride = 1920 elems
    g1[6] = 0;                                   // stride hi | dim1_stride lo
    g1[7] = 0;                                   // dim1_stride hi (unused, 2D)

    v4i z4; z4[0] = 0; z4[1] = 0; z4[2] = 0; z4[3] = 0;
    v8i z8; z8[0] = 0; z8[1] = 0; z8[2] = 0; z8[3] = 0;
    z8[4] = 0; z8[5] = 0; z8[6] = 0; z8[7] = 0;
    __builtin_amdgcn_tensor_load_to_lds(g0, g1, z4, z4, z8, 0);
    __builtin_amdgcn_s_wait_tensorcnt(0);
  }
  __syncthreads();

  // ---- resample: fold transforms into the sampling coordinate -----------
  const float scale = 128.0f / 300.0f;
  for (int pix = threadIdx.x; pix < PH * PW; pix += blockDim.x) {
    int oy = pix / PW, ox = pix % PW;
    float fy = ((float)oy + 0.5f) * scale - 0.5f;
    float fx = ((float)ox + 0.5f) * scale - 0.5f;
    // peel transforms outermost-first: rot90k3, then flipLR, then flipUD
    float Y = fy, X = fx;
    if (dec0) { float t = Y; Y = 127.f - X; X = t; }
    if (dec1) { X = 127.f - X; }
    if (dec2) { Y = 127.f - Y; }
    Y = fminf(fmaxf(Y, 0.f), 127.f);
    X = fminf(fmaxf(X, 0.f), 127.f);
    int y0 = (int)Y; if (y0 > 126) y0 = 126;
    int x0 = (int)X; if (x0 > 126) x0 = 126;
    float wy = Y - (float)y0;
    float wx = X - (float)x0;

    const float* t00 = &tile[(y0 * CROP + x0) * CCH];
    const float* t01 = t00 + CCH;
    const float* t10 = t00 + CROP * CCH;
    const float* t11 = t10 + CCH;
    float w00 = (1.f - wy) * (1.f - wx);
    float w01 = (1.f - wy) * wx;
    float w10 = wy * (1.f - wx);
    float w11 = wy * wx;

    float* o = &out[pix * CCH];
#pragma unroll
    for (int c = 0; c < CCH; ++c) {
      float v = w00 * t00[c] + w01 * t01[c] + w10 * t10[c] + w11 * t11[c];
      atomicAdd(&o[c], v);
    }
  }
}

// ---------------------------------------------------------------------------
extern "C" void kernel_launch(void* const* d_in, const int* in_sizes, int n_in,
                              void* d_out, int out_size, void* d_ws, size_t ws_size,
                              hipStream_t stream) {
  const float*         grads = (const float*)d_in[0];         // [8,640,640,3] f32
  const float*         patch = (const float*)d_in[1];         // [300,300,3]   f32
  const int*           boxes = (const int*)d_in[2];           // [8,16,4]      i32
  const unsigned char* decs  = (const unsigned char*)d_in[3]; // [8,16,3]      bool
  float*               out   = (float*)d_out;                 // [300,300,3]   f32

  int n = PH * PW * CCH;
  tv_grad_kernel<<<(n + 255) / 256, 256, 0, stream>>>(patch, out);

  patch_accum_kernel<<<NB * NP, 1024, CROP * CROP * CCH * sizeof(float), stream>>>(
      grads, boxes, decs, out);
}